// JointNetwork_24481313587346
// MI455X (gfx1250) — compile-verified
//
#include <hip/hip_runtime.h>
#include <cstddef>

typedef float v2f __attribute__((ext_vector_type(2)));
typedef float v8f __attribute__((ext_vector_type(8)));

namespace jn {
constexpr int KD  = 512;    // D
constexpr int NV  = 1024;   // V
constexpr int LDW = 1024;   // W row stride (2*D)
constexpr int BB  = 4;
constexpr int TT  = 200;
constexpr int UU  = 50;
}

// ---------------------------------------------------------------------------
// Kernel 1: elementwise tanh into workspace
// ---------------------------------------------------------------------------
__global__ __launch_bounds__(256)
void jn_tanh_kernel(const float* __restrict__ src, float* __restrict__ dst, int n) {
    int i = blockIdx.x * blockDim.x + threadIdx.x;
    if (i < n) dst[i] = tanhf(src[i]);
}

// ---------------------------------------------------------------------------
// Kernel 2: C[m0:m0+16, v0:v0+64] = A[m0:m0+16, :] @ W[v, col_off:+512]^T (+b)
// Full-precision f32 WMMA 16x16x4, register-blocked 4x in N:
// per K-step: 1 A load + 4 B loads -> 4 WMMAs (vs 2 loads : 1 WMMA before).
// C rows are padded to a multiple of 16 in the workspace so every store is
// unconditional (no exec-mask ladder in the epilogue).
// A (16x4): lane<16 -> row=lane&15, K={0,1}; lane>=16 -> K={2,3}.
// B (4x16): symmetric in N. C (16x16): VGPR r -> row r (lanes 0-15) /
// row r+8 (lanes 16-31), col = lane&15.
// ---------------------------------------------------------------------------
template <bool HAS_BIAS>
__global__ __launch_bounds__(128)
void jn_gemm_wmma_kernel(const float* __restrict__ A,   // M x 512 (already tanh'd)
                         const float* __restrict__ W,   // 1024 x 1024 row-major
                         int col_off,                   // 0 for We, 512 for Wp
                         const float* __restrict__ bias,
                         float* __restrict__ C,         // Mpad x 1024 (ws, padded)
                         int M) {
    using namespace jn;
    const int lane = threadIdx.x & 31;
    const int wave = threadIdx.x >> 5;
    const int m0   = blockIdx.x * 16;
    const int v0   = (blockIdx.y * 4 + wave) * 64;   // 64 output cols per wave
    const int hi   = lane >> 4;                      // 0: K {0,1}, 1: K {2,3}
    const int l15  = lane & 15;

    int mA = m0 + l15;                 // clamp loads: EXEC stays all-1s for WMMA
    if (mA > M - 1) mA = M - 1;
    const float* aptr = A + (size_t)mA * KD + (hi << 1);
    const float* b0   = W + (size_t)(v0 + l15) * LDW + col_off + (hi << 1);
    // 4 B tiles are 16 rows of W apart:
    const size_t bstep = (size_t)16 * LDW;

    v8f acc0 = {0.f,0.f,0.f,0.f,0.f,0.f,0.f,0.f};
    v8f acc1 = acc0, acc2 = acc0, acc3 = acc0;

    #pragma unroll 4
    for (int k = 0; k < KD; k += 4) {
        const v2f a  = *(const v2f*)(aptr + k);              // global_load_b64
        const v2f q0 = *(const v2f*)(b0 + k);
        const v2f q1 = *(const v2f*)(b0 + bstep + k);
        const v2f q2 = *(const v2f*)(b0 + 2 * bstep + k);
        const v2f q3 = *(const v2f*)(b0 + 3 * bstep + k);
        acc0 = __builtin_amdgcn_wmma_f32_16x16x4_f32(false, a, false, q0,
                                                     (short)0, acc0, false, false);
        acc1 = __builtin_amdgcn_wmma_f32_16x16x4_f32(false, a, false, q1,
                                                     (short)0, acc1, false, false);
        acc2 = __builtin_amdgcn_wmma_f32_16x16x4_f32(false, a, false, q2,
                                                     (short)0, acc2, false, false);
        acc3 = __builtin_amdgcn_wmma_f32_16x16x4_f32(false, a, false, q3,
                                                     (short)0, acc3, false, false);
    }

    float bv0 = 0.f, bv1 = 0.f, bv2 = 0.f, bv3 = 0.f;
    if (HAS_BIAS) {
        bv0 = bias[v0 + l15];
        bv1 = bias[v0 + 16 + l15];
        bv2 = bias[v0 + 32 + l15];
        bv3 = bias[v0 + 48 + l15];
    }

    float* crow = C + (size_t)(m0 + (hi << 3)) * NV + v0 + l15;
    #pragma unroll
    for (int r = 0; r < 8; ++r) {        // unconditional: C rows are padded
        float* cr = crow + (size_t)r * NV;
        cr[0]  = acc0[r] + bv0;
        cr[16] = acc1[r] + bv1;
        cr[32] = acc2[r] + bv2;
        cr[48] = acc3[r] + bv3;
    }
}

// ---------------------------------------------------------------------------
// Kernel 3: out[b,t,u,:] = log_softmax(ep[b,t,:] + pp[b,u,:])
// One 256-thread block per (b,t,u); 4 floats/thread; ep/pp stream from L2.
// ---------------------------------------------------------------------------
__global__ __launch_bounds__(256)
void jn_logsoftmax_kernel(const float* __restrict__ ep,   // (B*T) x V, bias included
                          const float* __restrict__ pp,   // (B*U) x V (padded rows unused)
                          float* __restrict__ out) {      // (B*T*U) x V
    using namespace jn;
    const int r = blockIdx.x;                 // flat (b,t,u)
    const int u = r % UU;
    const int t = (r / UU) % TT;
    const int b = r / (UU * TT);

    const float* eprow = ep  + (size_t)(b * TT + t) * NV;
    const float* pprow = pp  + (size_t)(b * UU + u) * NV;
    float*       orow  = out + (size_t)r * NV;

    const int tid = threadIdx.x;
    const int w   = tid >> 5;
    const int ln  = tid & 31;

    const float4 e4 = ((const float4*)eprow)[tid];
    const float4 p4 = ((const float4*)pprow)[tid];
    const float x0 = e4.x + p4.x, x1 = e4.y + p4.y;
    const float x2 = e4.z + p4.z, x3 = e4.w + p4.w;

    __shared__ float smax[8];
    __shared__ float ssum[8];

    // ---- block max ----
    float m = fmaxf(fmaxf(x0, x1), fmaxf(x2, x3));
    #pragma unroll
    for (int o = 16; o > 0; o >>= 1) m = fmaxf(m, __shfl_xor(m, o, 32));
    if (ln == 0) smax[w] = m;
    __syncthreads();
    if (w == 0) {
        float mm = (ln < 8) ? smax[ln] : -3.4e38f;
        #pragma unroll
        for (int o = 4; o > 0; o >>= 1) mm = fmaxf(mm, __shfl_xor(mm, o, 32));
        if (ln == 0) smax[0] = mm;
    }
    __syncthreads();
    m = smax[0];

    // ---- block sum of exp ----
    float s = __expf(x0 - m) + __expf(x1 - m) + __expf(x2 - m) + __expf(x3 - m);
    #pragma unroll
    for (int o = 16; o > 0; o >>= 1) s += __shfl_xor(s, o, 32);
    if (ln == 0) ssum[w] = s;
    __syncthreads();
    if (w == 0) {
        float ss = (ln < 8) ? ssum[ln] : 0.f;
        #pragma unroll
        for (int o = 4; o > 0; o >>= 1) ss += __shfl_xor(ss, o, 32);
        if (ln == 0) ssum[0] = ss;
    }
    __syncthreads();

    const float lse = m + __logf(ssum[0]);
    ((float4*)orow)[tid] = make_float4(x0 - lse, x1 - lse, x2 - lse, x3 - lse);
}

// ---------------------------------------------------------------------------
// kernel_launch
// ---------------------------------------------------------------------------
extern "C" void kernel_launch(void* const* d_in, const int* in_sizes, int n_in,
                              void* d_out, int out_size, void* d_ws, size_t ws_size,
                              hipStream_t stream) {
    using namespace jn;
    const float* enc  = (const float*)d_in[0];  // (4,200,512)
    const float* pred = (const float*)d_in[1];  // (4,50,512)
    const float* W    = (const float*)d_in[2];  // (1024,1024)
    const float* bias = (const float*)d_in[3];  // (1024,)

    const int Me  = BB * TT;             // 800 (multiple of 16)
    const int Mp  = BB * UU;             // 200
    const int MpP = ((Mp + 15) / 16) * 16; // 208: padded rows for guard-free stores

    // workspace layout (floats): Ae | Ap | ep | pp  => ~6.2 MB
    float* ws = (float*)d_ws;
    float* Ae = ws;
    float* Ap = Ae + (size_t)Me * KD;
    float* ep = Ap + (size_t)Mp * KD;
    float* pp = ep + (size_t)Me * NV;    // pp has MpP*NV floats of room

    jn_tanh_kernel<<<(Me * KD + 255) / 256, 256, 0, stream>>>(enc, Ae, Me * KD);
    jn_tanh_kernel<<<(Mp * KD + 255) / 256, 256, 0, stream>>>(pred, Ap, Mp * KD);

    // ep = tanh(enc) @ We^T + b   (We = W[:, 0:512])
    jn_gemm_wmma_kernel<true><<<dim3(Me / 16, NV / 256), 128, 0, stream>>>(
        Ae, W, /*col_off=*/0, bias, ep, Me);
    // pp = tanh(pred) @ Wp^T      (Wp = W[:, 512:1024]); 208 padded rows
    jn_gemm_wmma_kernel<false><<<dim3(MpP / 16, NV / 256), 128, 0, stream>>>(
        Ap, W, /*col_off=*/KD, nullptr, pp, Mp);

    // fused broadcast-add + log_softmax over 40000 rows of 1024
    jn_logsoftmax_kernel<<<BB * TT * UU, 256, 0, stream>>>(ep, pp, (float*)d_out);
}